// EmbeddingBag_59682865545864
// MI455X (gfx1250) — compile-verified
//
#include <hip/hip_runtime.h>

// ---------------------------------------------------------------------------
// EmbeddingBag(sum, per_sample_weights):
//   out[b, :] = sum_i  w[start_b + i] * weight[inp[start_b + i], :]
//
// CDNA5 async-gather kernel (gfx1250).
//
// One wave32 per bag. Each 128-float (512 B) embedding row is gathered
// HBM -> LDS with a single GLOBAL_LOAD_ASYNC_TO_LDS_B128 (32 lanes x 16 B),
// tracked on ASYNCcnt (in-order completion). Per-wave LDS ring:
//   SLOTS = 16 rows, DEPTH = 8 rows outstanding.
// Rolling pipeline:
//   prologue: issue DEPTH async row-gathers
//   steady:   preload idx[i+DEPTH], w[i] (scalar path, overlaps waits)
//             s_wait_asynccnt DEPTH-1   -> row i complete in LDS
//             ds_load_b128 row i, 4x fma, issue row i+DEPTH into slot
//             (i+DEPTH)&15  (8 iterations away from its last reader: no WAR)
//   tail:     s_wait_asynccnt 0, drain
// In-flight row data costs zero VGPRs; index/weight streams stay on the
// scalar path (uniform bag id via readfirstlane -> s_load), so the async op
// uses GVS addressing: saddr holds the row base in SGPRs, vaddr holds lane*16 in a VGPR,
// vdst holds the LDS byte address in a VGPR.
//
// Roofline: ~434 MB of traffic, ~18.6 us floor at 23.3 TB/s. FLOPs are
// negligible (210 MFLOP) and each bag gathers distinct rows, so there is no
// shared-operand matmul structure: WMMA would be pure overhead here.
// ---------------------------------------------------------------------------

#define D_EMB 128
#define ROW_BYTES (D_EMB * 4)
#define SLOTS 16               // LDS ring slots per wave (8 KB)
#define DEPTH 8                // async rows outstanding per wave
#define WAVES_PER_BLOCK 8

__global__ __launch_bounds__(WAVES_PER_BLOCK * 32, 1)
void embag_async_kernel(const int* __restrict__ inp,
                        const int* __restrict__ offsets,
                        const float* __restrict__ psw,
                        const float* __restrict__ weight,
                        float* __restrict__ out,
                        int n_bags, int n_idx) {
  __shared__ float ring[WAVES_PER_BLOCK][SLOTS][D_EMB];  // 64 KB / block

  const int lane = threadIdx.x & 31;
  const int wv   = threadIdx.x >> 5;
  // Uniform bag id -> scalar path for all per-bag metadata.
  const int bag = __builtin_amdgcn_readfirstlane(
      (int)(blockIdx.x * WAVES_PER_BLOCK + wv));
  if (bag >= n_bags) return;

  const int start = offsets[bag];
  const int end   = (bag + 1 < n_bags) ? offsets[bag + 1] : n_idx;
  const int count = end - start;

  const int*   bi = inp + start;
  const float* bw = psw + start;
  __builtin_prefetch(bi, 0, 0);   // global_prefetch_b8: warm index stream
  __builtin_prefetch(bw, 0, 0);

  const unsigned voff = (unsigned)lane * 16u;  // lane's 16 B slice of a row
  // Low 32 bits of the generic pointer == DS address on CDNA5.
  const unsigned lds_base = (unsigned)(uintptr_t)(&ring[wv][0][0]);
  const unsigned long long wbase = (unsigned long long)(uintptr_t)weight;

  // Issue one async row gather: HBM row(idx) -> ring slot (i & (SLOTS-1)).
  auto issue_row = [&](int i, unsigned idx) {
    const unsigned long long rowbase =
        wbase + (unsigned long long)idx * (unsigned long long)ROW_BYTES;
    const unsigned lds_addr =
        lds_base + (unsigned)(i & (SLOTS - 1)) * (unsigned)ROW_BYTES + voff;
    // GVS mode: copies 16 B from the global row base in saddr (SGPR pair) plus offset vaddr to the LDS address in vdst
    asm volatile("global_load_async_to_lds_b128 %0, %1, %2"
                 :
                 : "v"(lds_addr), "v"(voff), "s"(rowbase)
                 : "memory");
  };

  float4 acc = make_float4(0.0f, 0.0f, 0.0f, 0.0f);

  // Consume row i from the ring (its async load must be complete).
  auto consume_row = [&](int i, float w) {
    const float4 r =
        *(const float4*)(&ring[wv][i & (SLOTS - 1)][lane * 4]);  // ds_load_b128
    acc.x = fmaf(w, r.x, acc.x);
    acc.y = fmaf(w, r.y, acc.y);
    acc.z = fmaf(w, r.z, acc.z);
    acc.w = fmaf(w, r.w, acc.w);
  };

  // ---- prologue: fill the pipeline ----
  const int pro = (count < DEPTH) ? count : DEPTH;
  for (int p = 0; p < pro; ++p) issue_row(p, (unsigned)bi[p]);

  // ---- steady state: DEPTH rows outstanding ----
  int i = 0;
  for (; i + DEPTH < count; ++i) {
    // Preload scalars BEFORE the asynccnt wait so the KMcnt latency
    // overlaps the async wait window (the asm memory clobbers otherwise
    // pin these s_loads inside the critical path).
    const unsigned idx_n = (unsigned)bi[i + DEPTH];  // s_load
    const float    w_c   = bw[i];                    // s_load
    // In-order async completion: <= DEPTH-1 outstanding => row i is in LDS.
    asm volatile("s_wait_asynccnt 7" ::: "memory");
    consume_row(i, w_c);
    issue_row(i + DEPTH, idx_n);
  }

  // ---- tail: drain everything still in flight ----
  asm volatile("s_wait_asynccnt 0" ::: "memory");
  for (; i < count; ++i) consume_row(i, bw[i]);

  float4* o = (float4*)(out + (size_t)bag * D_EMB);
  o[lane] = acc;  // global_store_b128
}

// ---------------------------------------------------------------------------
// Host-side launcher.
// Inputs (setup_inputs order): [0] input int32 [819200], [1] offsets int32
// [16384], [2] per_sample_weights f32 [819200], [3] weight f32 [1e6*128].
// Output: f32 [16384*128].
// ---------------------------------------------------------------------------
extern "C" void kernel_launch(void* const* d_in, const int* in_sizes, int n_in,
                              void* d_out, int out_size, void* d_ws,
                              size_t ws_size, hipStream_t stream) {
  const int*   inp     = (const int*)d_in[0];
  const int*   offsets = (const int*)d_in[1];
  const float* psw     = (const float*)d_in[2];
  const float* weight  = (const float*)d_in[3];
  float*       out     = (float*)d_out;

  const int n_idx  = in_sizes[0];
  const int n_bags = in_sizes[1];

  const int blocks = (n_bags + WAVES_PER_BLOCK - 1) / WAVES_PER_BLOCK;
  embag_async_kernel<<<blocks, WAVES_PER_BLOCK * 32, 0, stream>>>(
      inp, offsets, psw, weight, out, n_bags, n_idx);
}